// YoloV1Loss_88656714924108
// MI455X (gfx1250) — compile-verified
//
#include <hip/hip_runtime.h>
#include <hip/hip_bf16.h>

#define CH 30
#define CELLS_PER_BLOCK 256
#define FLOATS_PER_BLOCK (CELLS_PER_BLOCK * CH)   // 7680 floats = 30720 B per tensor
#define CHUNKS (FLOATS_PER_BLOCK / 4)             // 1920 x 16B chunks per tensor

// ---- CDNA5 async global->LDS path (probe-verified in rounds 1-2) ----
#if defined(__has_builtin)
#  if __has_builtin(__builtin_amdgcn_global_load_async_to_lds_b128)
#    define HAVE_ASYNC_COPY 1
#  else
#    define HAVE_ASYNC_COPY 0
#    pragma message("probe: __builtin_amdgcn_global_load_async_to_lds_b128 NOT available; using fallback loads")
#  endif
#  if __has_builtin(__builtin_amdgcn_s_wait_asynccnt)
#    define HAVE_WAIT_ASYNC 1
#  else
#    define HAVE_WAIT_ASYNC 0
#    pragma message("probe: __builtin_amdgcn_s_wait_asynccnt NOT available; using inline asm")
#  endif
#else
#  define HAVE_ASYNC_COPY 0
#  define HAVE_WAIT_ASYNC 0
#endif

// Builtin signature (probe-confirmed): (AS1 int4*, AS3 int4*, imm offset, imm cpol)
typedef int v4i __attribute__((ext_vector_type(4)));
typedef __attribute__((address_space(1))) v4i gv4i;
typedef __attribute__((address_space(3))) v4i lv4i;

// OFF is a byte offset applied to BOTH the global and LDS address
// -> one base, many imm offsets.
template <int OFF>
__device__ __forceinline__ void async_copy16(const float* gsrc, float* ldst) {
#if HAVE_ASYNC_COPY
  __builtin_amdgcn_global_load_async_to_lds_b128(
      (gv4i*)gsrc, (lv4i*)ldst, /*offset=*/OFF, /*cpol=*/0);
#else
  *(float4*)((char*)ldst + OFF) = *(const float4*)((const char*)gsrc + OFF);
#endif
}

__device__ __forceinline__ void wait_async_all() {
#if HAVE_WAIT_ASYNC
  __builtin_amdgcn_s_wait_asynccnt(0);
#else
  asm volatile("s_wait_asynccnt 0" ::: "memory");
#endif
}

__device__ __forceinline__ void corners7(float x, float y, float w, float h,
                                         float r, float c,
                                         float& x1, float& y1, float& x2, float& y2) {
  float cx = (x + c) / 7.0f;
  float cy = (y + r) / 7.0f;
  x1 = cx - w * 0.5f;  y1 = cy - h * 0.5f;
  x2 = cx + w * 0.5f;  y2 = cy + h * 0.5f;
}

__device__ __forceinline__ float iou8(float ax1, float ay1, float ax2, float ay2,
                                      float bx1, float by1, float bx2, float by2) {
  float iw = fmaxf(fminf(ax2, bx2) - fmaxf(ax1, bx1), 0.0f);
  float ih = fmaxf(fminf(ay2, by2) - fmaxf(ay1, by1), 0.0f);
  float inter = iw * ih;
  float u = (ax2 - ax1) * (ay2 - ay1) + (bx2 - bx1) * (by2 - by1) - inter;
  return (u > 0.0f) ? (inter / u) : 0.0f;
}

__global__ __launch_bounds__(CELLS_PER_BLOCK)
void yolo_loss_stage1(const float* __restrict__ data,
                      const float* __restrict__ labels,
                      float* __restrict__ partial,
                      int n_cells) {
  __shared__ float sD[FLOATS_PER_BLOCK];
  __shared__ float sL[FLOATS_PER_BLOCK];
  __shared__ float waveSum[CELLS_PER_BLOCK / 32];

  const int tid = threadIdx.x;
  const long long blockFloatBase = (long long)blockIdx.x * FLOATS_PER_BLOCK;
  const long long totalFloats    = (long long)n_cells * CH;

  // ---- Stage both tensors into LDS with coalesced 128-bit async copies ----
  {
    const float* gD = data   + blockFloatBase + (long long)tid * 4;
    const float* gL = labels + blockFloatBase + (long long)tid * 4;
    float* lD = &sD[tid * 4];
    float* lL = &sL[tid * 4];

    if (blockFloatBase + FLOATS_PER_BLOCK <= totalFloats) {
      // Full tile: 15 chunks/thread, pure immediate-offset async issue.
      async_copy16<0>(gD, lD);       async_copy16<0>(gL, lL);
      async_copy16<4096>(gD, lD);    async_copy16<4096>(gL, lL);
      async_copy16<8192>(gD, lD);    async_copy16<8192>(gL, lL);
      async_copy16<12288>(gD, lD);   async_copy16<12288>(gL, lL);
      async_copy16<16384>(gD, lD);   async_copy16<16384>(gL, lL);
      async_copy16<20480>(gD, lD);   async_copy16<20480>(gL, lL);
      async_copy16<24576>(gD, lD);   async_copy16<24576>(gL, lL);
      if (tid < 128) {               // chunks 1792..1919
        async_copy16<28672>(gD, lD); async_copy16<28672>(gL, lL);
      }
    } else {
      // Generic guarded tail path (unused for exact-fit shapes, kept for safety).
      for (int i = tid; i < CHUNKS; i += CELLS_PER_BLOCK) {
        long long g = blockFloatBase + (long long)i * 4;
        if (g + 3 < totalFloats) {
          async_copy16<0>(data + g,   &sD[i * 4]);
          async_copy16<0>(labels + g, &sL[i * 4]);
        }
      }
    }
  }
#if HAVE_ASYNC_COPY
  wait_async_all();          // this wave's ASYNCcnt -> 0 (LDS writes landed)
#endif
  __syncthreads();           // all waves' staging visible

  const int n = blockIdx.x * CELLS_PER_BLOCK + tid;   // global cell index
  float loss = 0.0f;
  if (n < n_cells) {
    const float* D = &sD[tid * CH];                   // stride 30 dwords: bank-conflict-free
    const float* L = &sL[tid * CH];
    const int cell = n % 49;
    const float r = (float)(cell / 7);
    const float c = (float)(cell % 7);

    float b1x1, b1y1, b1x2, b1y2, b2x1, b2y1, b2x2, b2y2, gx1, gy1, gx2, gy2;
    corners7(D[0], D[1], D[2], D[3], r, c, b1x1, b1y1, b1x2, b1y2);
    corners7(D[5], D[6], D[7], D[8], r, c, b2x1, b2y1, b2x2, b2y2);
    corners7(L[0], L[1], L[2], L[3], r, c, gx1, gy1, gx2, gy2);

    const float iou1 = iou8(b1x1, b1y1, b1x2, b1y2, gx1, gy1, gx2, gy2);
    const float iou2 = iou8(b2x1, b2y1, b2x2, b2y2, gx1, gy1, gx2, gy2);
    const bool resp1 = (iou1 >= iou2);

    float t0 = D[0] - L[0], t1 = D[1] - L[1];
    const float xy1 = t0 * t0 + t1 * t1;
    float t5 = D[5] - L[5], t6 = D[6] - L[6];
    const float xy2 = t5 * t5 + t6 * t6;

    float s2 = sqrtf(D[2]) - sqrtf(L[2]), s3 = sqrtf(D[3]) - sqrtf(L[3]);
    const float wh1 = s2 * s2 + s3 * s3;
    float s7 = sqrtf(D[7]) - sqrtf(L[7]), s8 = sqrtf(D[8]) - sqrtf(L[8]);
    const float wh2 = s7 * s7 + s8 * s8;

    const float d4 = D[4], d9 = D[9];
    const float co = 5.0f * (resp1 ? xy1 : xy2);
    const float wh = 5.0f * (resp1 ? wh1 : wh2);
    const float c1 = d4 - iou1, c2 = d9 - iou2;
    const float confi    = resp1 ? (c1 * c1) : (c2 * c2);
    const float noobj_in = 0.5f * (resp1 ? (d9 * d9) : (d4 * d4));

    float cls = 0.0f;
#pragma unroll
    for (int k = 10; k < CH; ++k) {
      float dk = D[k] - L[k];
      cls += dk * dk;
    }

    const float m = (L[4] == 1.0f) ? 1.0f : 0.0f;
    loss = m * (co + wh + confi + noobj_in + cls)
         + (1.0f - m) * 0.5f * (d4 * d4 + d9 * d9);
  }

  // wave32 butterfly reduction
#pragma unroll
  for (int off = 16; off > 0; off >>= 1)
    loss += __shfl_xor(loss, off, 32);
  if ((tid & 31) == 0) waveSum[tid >> 5] = loss;
  __syncthreads();
  if (tid == 0) {
    float s = 0.0f;
#pragma unroll
    for (int w = 0; w < CELLS_PER_BLOCK / 32; ++w) s += waveSum[w];
    partial[blockIdx.x] = s;
  }
}

__global__ __launch_bounds__(256)
void yolo_loss_reduce(const float* __restrict__ partial, int nparts,
                      float invB, float* __restrict__ out) {
  __shared__ float waveSum[8];
  float s = 0.0f;
  for (int i = threadIdx.x; i < nparts; i += 256) s += partial[i];
#pragma unroll
  for (int off = 16; off > 0; off >>= 1)
    s += __shfl_xor(s, off, 32);
  if ((threadIdx.x & 31) == 0) waveSum[threadIdx.x >> 5] = s;
  __syncthreads();
  if (threadIdx.x == 0) {
    float t = 0.0f;
#pragma unroll
    for (int w = 0; w < 8; ++w) t += waveSum[w];
    out[0] = t * invB;     // overwrite (no atomics): deterministic across replays
  }
}

extern "C" void kernel_launch(void* const* d_in, const int* in_sizes, int n_in,
                              void* d_out, int out_size, void* d_ws, size_t ws_size,
                              hipStream_t stream) {
  const float* data   = (const float*)d_in[0];
  const float* labels = (const float*)d_in[1];
  float* out     = (float*)d_out;
  float* partial = (float*)d_ws;   // needs (n_cells/256) floats ≈ 6.3 KB

  const long long totalFloats = (long long)in_sizes[0];
  const int n_cells = (int)(totalFloats / CH);       // 8192*49 = 401408
  const int B       = n_cells / 49;                  // 8192
  const int blocks  = (n_cells + CELLS_PER_BLOCK - 1) / CELLS_PER_BLOCK;  // 1568

  yolo_loss_stage1<<<blocks, CELLS_PER_BLOCK, 0, stream>>>(data, labels, partial, n_cells);
  yolo_loss_reduce<<<1, 256, 0, stream>>>(partial, blocks, 1.0f / (float)B, out);
}